// TAM_14929306320962
// MI455X (gfx1250) — compile-verified
//
#include <hip/hip_runtime.h>
#include <math.h>

#define BB 32
#define CC 256
#define HH 64
#define WW 64
#define HW 4096

typedef __attribute__((ext_vector_type(2))) float v2f;
typedef __attribute__((ext_vector_type(8))) float v8f;

// ---------------------------------------------------------------- zero scratch
__global__ __launch_bounds__(256) void TAM_zero_kernel(float* p, int n) {
  int i = blockIdx.x * 256 + threadIdx.x;
  if (i < n) p[i] = 0.f;
}

// ---------------------------------------------------------------- pass 1: means
// One block per (b,c) plane (4096 floats). Produces:
//   xbar1[b,c]      = mean over (h,w)         (direct store, pre-scaled 1/4096)
//   xbar2acc[b,h]  += sum_w x[b,c,h,w]        (atomic over c; scaled later)
//   xbar3acc[b,w]  += sum_h x[b,c,h,w]        (atomic over c; scaled later)
__global__ __launch_bounds__(256)
void TAM_means_kernel(const float* __restrict__ x, float* __restrict__ xbar1,
                      float* __restrict__ xbar2acc, float* __restrict__ xbar3acc) {
  const int c = blockIdx.x, b = blockIdx.y, t = threadIdx.x;
  const float* plane = x + ((size_t)(b * CC + c)) * HW;
  __shared__ float red[256];

  // chunked pattern: thread t -> row r=t/4, quarter q=t%4 (16 contiguous floats)
  const int r = t >> 2, q = t & 3;
  const float4* p4 = (const float4*)(plane + r * WW + q * 16);
  float4 a0 = p4[0], a1 = p4[1], a2 = p4[2], a3 = p4[3];
  float rpart = (a0.x + a0.y + a0.z + a0.w) + (a1.x + a1.y + a1.z + a1.w) +
                (a2.x + a2.y + a2.z + a2.w) + (a3.x + a3.y + a3.z + a3.w);
  red[t] = rpart;
  __syncthreads();
  if (t < 64) {
    float rs = red[4 * t] + red[4 * t + 1] + red[4 * t + 2] + red[4 * t + 3];
    atomicAdd(&xbar2acc[b * HH + t], rs);          // row h = t
  }
  __syncthreads();
  for (int s = 128; s > 0; s >>= 1) {
    if (t < s) red[t] += red[t + s];
    __syncthreads();
  }
  if (t == 0) xbar1[b * CC + c] = red[0] * (1.0f / HW);
  __syncthreads();

  // strided pattern: thread t -> fixed w=t%64, h = t/64 + 4i
  const int w = t & 63, h0 = t >> 6;
  float cpart = 0.f;
#pragma unroll
  for (int i = 0; i < 16; ++i) cpart += plane[(h0 + 4 * i) * WW + w];
  red[t] = cpart;
  __syncthreads();
  if (t < 64) {
    float cs = red[t] + red[t + 64] + red[t + 128] + red[t + 192];
    atomicAdd(&xbar3acc[b * WW + t], cs);          // col w = t
  }
}

// ------------------------------------------------- pass 2: small WMMA f32 GEMMs
// C[MxN] = alpha * A[MxK] * B ; B row-major [KxN] (TRANS_B=0) or accessed as
// B[n][k] from a row-major [NxK] array (TRANS_B=1). One wave per 16x16 tile.
// grid.x = N/16, grid.y = M/16. Uses V_WMMA_F32_16X16X4_F32 (CDNA5, wave32).
// TRANS_B is a template parameter so each specialization is a straight-line
// K-loop (no per-iteration branches; loads can be clustered ahead of v_wmma).
template <int TRANS_B>
__global__ __launch_bounds__(32)
void TAM_wmma_gemm_f32(const float* __restrict__ A, const float* __restrict__ B,
                       float* __restrict__ Cm, int N, int K, float alpha) {
  const int tn = blockIdx.x * 16, tm = blockIdx.y * 16;
  const int lane = threadIdx.x;
  const int half = lane >> 4;     // 0: K={0,1}; 1: K={2,3} (ISA A/B fragment split)
  const int l = lane & 15;

  // Loop-invariant per-lane base pointers; inner loop just strides by K.
  const float* ap = A + (size_t)(tm + l) * K + 2 * half;
  const float* bp = TRANS_B ? (B + (size_t)(tn + l) * K + 2 * half)
                            : (B + (size_t)(2 * half) * N + (tn + l));

  v8f acc = {};
  for (int k0 = 0; k0 < K; k0 += 4) {
    v2f a, bfrag;
    a.x = ap[0];
    a.y = ap[1];
    if (TRANS_B) {
      bfrag.x = bp[0];
      bfrag.y = bp[1];
    } else {
      bfrag.x = bp[0];
      bfrag.y = bp[N];
    }
    acc = __builtin_amdgcn_wmma_f32_16x16x4_f32(false, a, false, bfrag,
                                                (short)0, acc, false, false);
    ap += 4;
    bp += TRANS_B ? 4 : 4 * N;
  }
  // C/D layout: VGPR r holds row tm+r (lanes 0-15) / tm+8+r (lanes 16-31)
  const int row0 = tm + half * 8;
  const int col = tn + l;
#pragma unroll
  for (int r = 0; r < 8; ++r)
    Cm[(size_t)(row0 + r) * N + col] = alpha * acc[r];
}

// --------------------------------------------- pass 3: all three logit tensors
// One block per (b, group of 8 channels). Single read of x produces:
//   t1acc[b,hw] += u1[b,c]*x      (atomic across 32 groups; L2-resident 512 KB)
//   t2[b,c,w]    = sum_h u2[b,h]*x   (complete in-block)
//   t3[b,c,h]    = sum_w u3[b,w]*x   (complete in-block)
__global__ __launch_bounds__(256)
void TAM_logits_kernel(const float* __restrict__ x, const float* __restrict__ U1,
                       const float* __restrict__ U2, const float* __restrict__ U3,
                       float* __restrict__ t1acc, float* __restrict__ t2,
                       float* __restrict__ t3) {
  const int g = blockIdx.x;          // channel group (8 channels)
  const int b = blockIdx.y;
  const int t = threadIdx.x;
  __shared__ float su2[64], su3[64], red[256];
  if (t < 64) { su2[t] = U2[b * HH + t]; su3[t] = U3[b * WW + t]; }
  __syncthreads();

  float acc1[16];
#pragma unroll
  for (int i = 0; i < 16; ++i) acc1[i] = 0.f;

  const int h0 = t >> 6;                 // strided pattern (w = t & 63 fixed)
  const int r = t >> 2, q = t & 3;       // chunked pattern

  for (int j = 0; j < 8; ++j) {
    const int c = g * 8 + j;
    const float* plane = x + ((size_t)(b * CC + c)) * HW;
    const float u1c = U1[b * CC + c];

    // strided sweep: t1 accumulation (fixed pixel set per thread) + t2 partials
    float p2 = 0.f;
#pragma unroll
    for (int i = 0; i < 16; ++i) {
      float v = plane[t + 256 * i];
      acc1[i] += u1c * v;
      p2 += su2[h0 + 4 * i] * v;
    }
    red[t] = p2;
    __syncthreads();
    if (t < 64)
      t2[((size_t)b * CC + c) * WW + t] =
          red[t] + red[t + 64] + red[t + 128] + red[t + 192];
    __syncthreads();

    // chunked sweep (L2/WGP$-hot re-read): t3 partials
    float p3 = 0.f;
    const float* rowp = plane + r * WW + q * 16;
#pragma unroll
    for (int i = 0; i < 16; ++i) p3 += su3[q * 16 + i] * rowp[i];
    red[t] = p3;
    __syncthreads();
    if (t < 64)
      t3[((size_t)b * CC + c) * HH + t] =
          red[4 * t] + red[4 * t + 1] + red[4 * t + 2] + red[4 * t + 3];
    __syncthreads();
  }
#pragma unroll
  for (int i = 0; i < 16; ++i)
    atomicAdd(&t1acc[(size_t)b * HW + t + 256 * i], acc1[i]);
}

// ------------------------------------- pass 4: softmax -> sigmoid, in place
__global__ __launch_bounds__(256)
void TAM_softmax_sig_kernel(float* __restrict__ t1acc, float* __restrict__ t2,
                            float* __restrict__ t3) {
  const int br = blockIdx.x, b = blockIdx.y, t = threadIdx.x;
  float* p; int n;
  if (br == 0)      { p = t1acc + (size_t)b * HW;    n = HW;    }
  else if (br == 1) { p = t2 + (size_t)b * CC * WW;  n = CC*WW; }
  else              { p = t3 + (size_t)b * CC * HH;  n = CC*HH; }
  __shared__ float red[256];

  float m = -INFINITY;
  for (int i = t; i < n; i += 256) m = fmaxf(m, p[i]);
  red[t] = m; __syncthreads();
  for (int s = 128; s > 0; s >>= 1) {
    if (t < s) red[t] = fmaxf(red[t], red[t + s]);
    __syncthreads();
  }
  m = red[0]; __syncthreads();

  float zs = 0.f;
  for (int i = t; i < n; i += 256) zs += expf(p[i] - m);
  red[t] = zs; __syncthreads();
  for (int s = 128; s > 0; s >>= 1) {
    if (t < s) red[t] += red[t + s];
    __syncthreads();
  }
  const float inv = 1.0f / red[0];

  for (int i = t; i < n; i += 256) {
    float sm = expf(p[i] - m) * inv;
    p[i] = 1.0f / (1.0f + expf(-sm));      // sigmoid(softmax)
  }
}

// -------------------------- pass 5: out = x * (s1[hw] + s2[c,w] + s3[c,h])
__global__ __launch_bounds__(256)
void TAM_apply_kernel(const float* __restrict__ x, const float* __restrict__ s1,
                      const float* __restrict__ s2, const float* __restrict__ s3,
                      float* __restrict__ out) {
  const int c = blockIdx.x, b = blockIdx.y, t = threadIdx.x;
  const size_t base = ((size_t)(b * CC + c)) * HW;
  const int w = t & 63, h0 = t >> 6;
  const float sv2 = s2[((size_t)b * CC + c) * WW + w];
  const float* s3c = s3 + ((size_t)b * CC + c) * HH;
  const float* s1b = s1 + (size_t)b * HW;
#pragma unroll
  for (int i = 0; i < 16; ++i) {
    const int n = t + 256 * i;
    const int h = h0 + 4 * i;
    float v = x[base + n] * (s1b[n] + sv2 + s3c[h]);
    __builtin_nontemporal_store(v, &out[base + n]);  // don't evict L2-resident x
  }
}

// -------------------------------------------------------------------- launcher
extern "C" void kernel_launch(void* const* d_in, const int* in_sizes, int n_in,
                              void* d_out, int out_size, void* d_ws, size_t ws_size,
                              hipStream_t stream) {
  (void)in_sizes; (void)n_in; (void)out_size; (void)ws_size;
  const float* x   = (const float*)d_in[0];  // (32,256,64,64)
  const float* wl1 = (const float*)d_in[1];  // (128,256)
  const float* wl2 = (const float*)d_in[2];  // (128,256)
  const float* wr1 = (const float*)d_in[3];  // (32,64)
  const float* wr2 = (const float*)d_in[4];  // (32,64)

  float* ws = (float*)d_ws;
  float* xbar1 = ws;                  // 32*256   = 8192
  float* V1 = xbar1 + 8192;           // 32*128   = 4096
  float* U1 = V1 + 4096;              // 32*256   = 8192
  float* V2 = U1 + 8192;              // 32*32    = 1024
  float* U2 = V2 + 1024;              // 32*64    = 2048
  float* V3 = U2 + 2048;              // 32*32    = 1024
  float* U3 = V3 + 1024;              // 32*64    = 2048
  float* t2 = U3 + 2048;              // 32*256*64 = 524288
  float* t3 = t2 + 524288;            // 32*256*64 = 524288
  float* xbar2acc = t3 + 524288;      // 2048   } contiguous, zeroed together
  float* xbar3acc = xbar2acc + 2048;  // 2048   }
  float* t1acc = xbar3acc + 2048;     // 32*4096 = 131072 }  (~4.6 MiB total)

  const int nzero = 2048 + 2048 + 131072;
  TAM_zero_kernel<<<(nzero + 255) / 256, 256, 0, stream>>>(xbar2acc, nzero);

  TAM_means_kernel<<<dim3(CC, BB), 256, 0, stream>>>(x, xbar1, xbar2acc, xbar3acc);

  // u1 = wl2^T (wl1 xbar1) ; u2 = wr2^T (wr1 xbar2) ; u3 = wr2^T (wr1 xbar3)
  // (1/16384 mean divisor folded into the first GEMM of branches 2/3)
  TAM_wmma_gemm_f32<1><<<dim3(8, 2), 32, 0, stream>>>(xbar1, wl1, V1, 128, 256, 1.0f);
  TAM_wmma_gemm_f32<0><<<dim3(16, 2), 32, 0, stream>>>(V1, wl2, U1, 256, 128, 1.0f);
  TAM_wmma_gemm_f32<1><<<dim3(2, 2), 32, 0, stream>>>(xbar2acc, wr1, V2, 32, 64,
                                                      1.0f / 16384.0f);
  TAM_wmma_gemm_f32<0><<<dim3(4, 2), 32, 0, stream>>>(V2, wr2, U2, 64, 32, 1.0f);
  TAM_wmma_gemm_f32<1><<<dim3(2, 2), 32, 0, stream>>>(xbar3acc, wr1, V3, 32, 64,
                                                      1.0f / 16384.0f);
  TAM_wmma_gemm_f32<0><<<dim3(4, 2), 32, 0, stream>>>(V3, wr2, U3, 64, 32, 1.0f);

  TAM_logits_kernel<<<dim3(32, BB), 256, 0, stream>>>(x, U1, U2, U3, t1acc, t2, t3);
  TAM_softmax_sig_kernel<<<dim3(3, BB), 256, 0, stream>>>(t1acc, t2, t3);
  TAM_apply_kernel<<<dim3(CC, BB), 256, 0, stream>>>(x, t1acc, t2, t3, (float*)d_out);
}